// LocalCorrRatio_18631568130522
// MI455X (gfx1250) — compile-verified
//
#include <hip/hip_runtime.h>
#include <hip/hip_bf16.h>

// ---------------------------------------------------------------------------
// LocalCorrRatio loss, fully fused for MI455X (gfx1250, wave32).
//
//   out = -(1/12000) * sum over 4 calls x 1000 patches of eta^2(patch)
//
// One wave (32 threads) per (call, patch). Parzen weights computed with
// v_exp_f32 (exp2) on the TRANS pipe; per-bin reductions Sum(w), Sum(w*x)
// done with V_WMMA_F32_16X16X4_F32 (A = 16 bins x 4 voxels of weights,
// B = [ones | x] in columns 0..1), which co-executes with the exp stream.
// Deterministic two-stage reduction (no float atomics).
// ---------------------------------------------------------------------------

#define BINS    32
#define WIN     9
#define W3      729          // 9^3 voxels per patch
#define NPATCH  1000         // 10^3 patches per call
#define NCALLS  4
#define NBLK    (NCALLS * NPATCH)
#define DIM     90

typedef float v2f __attribute__((ext_vector_type(2)));
typedef float v8f __attribute__((ext_vector_type(8)));

__device__ __forceinline__ float fexp2(float x) {
#if __has_builtin(__builtin_amdgcn_exp2f)
  return __builtin_amdgcn_exp2f(x);   // v_exp_f32
#else
  return exp2f(x);
#endif
}

// PRETERM = 1/sigma^2 = 31^2 = 961 ;  exp(-961 d^2) = exp2(NEGK * d^2)
#define NEGK (-961.0f * 1.4426950408889634f)

__launch_bounds__(32)
__global__ void LocalCorrRatio_patch_kernel(const float* __restrict__ y_true,
                                            const float* __restrict__ y_pred,
                                            float* __restrict__ ws)
{
  __shared__ float xs[W3 + 4];      // X patch (intensities), zero-padded tail
  __shared__ float ys[W3 + 4];      // Y patch (histogrammed), zero-padded tail
  __shared__ float red[2][BINS];    // [0][b] = Sum w ; [1][b] = Sum w*x
  __shared__ float sxl[32], sxxl[32];

  const int bid  = blockIdx.x;           // 0 .. 3999
  const int lane = threadIdx.x;          // 0 .. 31
  const int call = bid / NPATCH;         // 0..3
  const int p    = bid - call * NPATCH;  // 0..999
  const int ph   = p / 100;
  const int pw   = (p / 10) % 10;
  const int pd   = p % 10;
  const int shift = (call >= 2) ? (WIN / 2) : 0;   // roll by -4 => +4 mod 90

  // call0: X=yt,Y=yp ; call1: X=yp,Y=yt ; call2/3 same with shift
  const float* __restrict__ Xsrc = (call & 1) ? y_pred : y_true;
  const float* __restrict__ Ysrc = (call & 1) ? y_true : y_pred;

  // ---- stage patch into LDS (rows of 9 contiguous floats) ----
  for (int j = lane; j < W3; j += 32) {
    int a = j / 81;
    int r = j - a * 81;
    int b = r / 9;
    int c = r - b * 9;
    int gh = ph * WIN + a + shift; if (gh >= DIM) gh -= DIM;
    int gw = pw * WIN + b + shift; if (gw >= DIM) gw -= DIM;
    int gd = pd * WIN + c + shift; if (gd >= DIM) gd -= DIM;
    int idx = (gh * DIM + gw) * DIM + gd;
    xs[j] = Xsrc[idx];
    ys[j] = Ysrc[idx];
  }
  if (lane < 4) { xs[W3 + lane] = 0.0f; ys[W3 + lane] = 0.0f; }
  __syncthreads();

  // ---- per-lane partial sums for patch mean / variance of X ----
  {
    float sx = 0.0f, sxx = 0.0f;
    for (int j = lane; j < W3; j += 32) {
      float v = xs[j];
      sx  += v;
      sxx += v * v;
    }
    sxl[lane]  = sx;
    sxxl[lane] = sxx;
  }

  // ---- WMMA bin reduction: C[b,0]=Sum w, C[b,1]=Sum w*x -------------------
  // A (16x4 f32, 2 VGPRs): lanes 0-15 hold M=lane, K=0(.x)/1(.y);
  //                        lanes 16-31 hold M=lane-16, K=2(.x)/3(.y).
  // B (4x16 f32, 2 VGPRs): same K striping, N = lane%16 within each row.
  const int   nIdx   = lane & 15;
  const int   hiHalf = lane >> 4;            // 0: K=0,1   1: K=2,3
  const int   kk0    = hiHalf * 2;
  const float cbL    = (float)nIdx * (1.0f / 31.0f);       // bins 0..15
  const float cbH    = cbL + (16.0f / 31.0f);              // bins 16..31

  v8f c0 = {};   // bins 0..15
  v8f c1 = {};   // bins 16..31

  for (int j0 = 0; j0 < W3; j0 += 4) {       // 183 iterations, EXEC uniform
    int jA = j0 + kk0;                        // this lane's two K voxels
    int jB = jA + 1;
    float m0 = (jA < W3) ? 1.0f : 0.0f;       // tail mask (729 = 4*182+1)
    float m1 = (jB < W3) ? 1.0f : 0.0f;
    float y0 = ys[jA];                        // pad reads return 0 (safe)
    float y1 = ys[jB];
    float x0 = xs[jA];
    float x1 = xs[jB];

    float d;
    v2f A0, A1, B;
    d = y0 - cbL; A0.x = m0 * fexp2(NEGK * d * d);
    d = y1 - cbL; A0.y = m1 * fexp2(NEGK * d * d);
    d = y0 - cbH; A1.x = m0 * fexp2(NEGK * d * d);
    d = y1 - cbH; A1.y = m1 * fexp2(NEGK * d * d);
    B.x = (nIdx == 0) ? 1.0f : ((nIdx == 1) ? x0 : 0.0f);
    B.y = (nIdx == 0) ? 1.0f : ((nIdx == 1) ? x1 : 0.0f);

    c0 = __builtin_amdgcn_wmma_f32_16x16x4_f32(false, A0, false, B,
                                               (short)0, c0, false, false);
    c1 = __builtin_amdgcn_wmma_f32_16x16x4_f32(false, A1, false, B,
                                               (short)0, c1, false, false);
  }

  // ---- dump C columns 0 (Sum w) and 1 (Sum w*x) to LDS --------------------
  // C layout: lane l, VGPR r -> row M = (l<16 ? r : r+8), col N = l%16.
  if (nIdx < 2) {
#pragma unroll
    for (int r = 0; r < 8; ++r) {
      int bin = hiHalf ? (r + 8) : r;
      red[nIdx][bin]      = c0[r];
      red[nIdx][bin + 16] = c1[r];
    }
  }
  __syncthreads();

  // ---- scalar finalize: eta^2 for this patch ------------------------------
  if (lane == 0) {
    float Sx = 0.0f, Sxx = 0.0f;
    for (int i = 0; i < 32; ++i) { Sx += sxl[i]; Sxx += sxxl[i]; }
    float mu = Sx * (1.0f / 729.0f);
    float tv = (Sxx - 729.0f * mu * mu) * (1.0f / 728.0f);   // ddof=1

    float sumSw = 0.0f, num = 0.0f;
    for (int b = 0; b < BINS; ++b) {
      float Sw  = red[0][b];
      float Swx = red[1][b];
      float m   = Swx / (Sw + 1e-5f);     // weights_norm mean intensity
      float dm  = m - mu;
      num   += Sw * dm * dm;
      sumSw += Sw;
    }
    float bgv = num / sumSw;
    ws[bid] = bgv / (tv + 1e-5f);         // eta^2
  }
}

// Deterministic final reduction: fixed-order strided sums + LDS tree.
__launch_bounds__(256)
__global__ void LocalCorrRatio_reduce_kernel(const float* __restrict__ ws,
                                             float* __restrict__ out)
{
  __shared__ float s[256];
  int t = threadIdx.x;
  float acc = 0.0f;
  for (int i = t; i < NBLK; i += 256) acc += ws[i];
  s[t] = acc;
  __syncthreads();
  for (int off = 128; off > 0; off >>= 1) {
    if (t < off) s[t] += s[t + off];
    __syncthreads();
  }
  // mean over patches, /3 per call, then -(CR)/4 - (CR_shifted)/4
  if (t == 0) out[0] = -s[0] * (1.0f / 12000.0f);
}

extern "C" void kernel_launch(void* const* d_in, const int* in_sizes, int n_in,
                              void* d_out, int out_size, void* d_ws, size_t ws_size,
                              hipStream_t stream)
{
  (void)in_sizes; (void)n_in; (void)out_size; (void)ws_size;
  const float* y_true = (const float*)d_in[0];
  const float* y_pred = (const float*)d_in[1];
  float* ws = (float*)d_ws;                 // 4000 floats of scratch

  LocalCorrRatio_patch_kernel<<<NBLK, 32, 0, stream>>>(y_true, y_pred, ws);
  LocalCorrRatio_reduce_kernel<<<1, 256, 0, stream>>>(ws, (float*)d_out);
}